// XLSTM_dynamic_graph_70317204570396
// MI455X (gfx1250) — compile-verified
//
#include <hip/hip_runtime.h>
#include <math.h>

// ---------------------------------------------------------------------------
// xLSTM-style cell scan for MI455X (gfx1250, wave32, WMMA).
// Dominant op: per (b,t) now = relu(LN(K @ V^T)), 1024x1024x128 fp32 GEMM,
// done with V_WMMA_F32_16X16X4_F32 (exact fp32, matches the fp32 reference).
// ---------------------------------------------------------------------------

typedef float v2f __attribute__((ext_vector_type(2)));
typedef float v8f __attribute__((ext_vector_type(8)));

#define BB 8
#define TT 16
#define NN 1024
#define FF 64
#define DD 128
#define DLL 8
#define LN_EPS 1e-5f

// ---------------------------------------------------------------------------
// Stage 0: label drift gate.  lab[b,0]=0 ; lab[b,t]=((||d||-mn)/(mx-mn))*2-1
// ---------------------------------------------------------------------------
__global__ void lab_kernel(const float* __restrict__ labels,
                           float* __restrict__ lab) {
  __shared__ float dnv[128], rmin[128], rmax[128];
  const int tid = threadIdx.x;
  const int NI = BB * (TT - 1); // 120 items
  float dn = 0.f;
  if (tid < NI) {
    const int b = tid / (TT - 1), j = tid % (TT - 1);
    const float* p0 = labels + (size_t)(b * TT + j) * DLL;
    const float* p1 = labels + (size_t)(b * TT + j + 1) * DLL;
    float s = 0.f;
#pragma unroll
    for (int d = 0; d < DLL; ++d) { float df = p1[d] - p0[d]; s += df * df; }
    dn = sqrtf(s);
  }
  dnv[tid] = dn;
  rmin[tid] = (tid < NI) ? dn : 3.4e38f;
  rmax[tid] = (tid < NI) ? dn : -3.4e38f;
  __syncthreads();
  for (int s = 64; s > 0; s >>= 1) {
    if (tid < s) {
      rmin[tid] = fminf(rmin[tid], rmin[tid + s]);
      rmax[tid] = fmaxf(rmax[tid], rmax[tid + s]);
    }
    __syncthreads();
  }
  const float mn = rmin[0], mx = rmax[0];
  if (tid < NI) {
    const int b = tid / (TT - 1), j = tid % (TT - 1);
    lab[b * TT + j + 1] = (dnv[tid] - mn) / (mx - mn) * 2.f - 1.f;
  } else if (tid < NI + BB) {
    lab[(tid - NI) * TT] = 0.f; // t = 0 entries
  }
}

// ---------------------------------------------------------------------------
// Stage 1 (per t): per-head K/V projection + LayerNorm over D + softplus,
// plus the scalar input gate.  One block per head n, 128 threads (= d).
// ---------------------------------------------------------------------------
__device__ __forceinline__ float softplus_f(float x) {
  return (x > 0.f) ? x + log1pf(expf(-x)) : log1pf(expf(x));
}

__global__ void kv_kernel(const float* __restrict__ x,
                          const float* __restrict__ Wk,
                          const float* __restrict__ Wv,
                          const float* __restrict__ bk,
                          const float* __restrict__ bv,
                          const float* __restrict__ w_in,
                          const float* __restrict__ b_in,
                          const float* __restrict__ kg,
                          const float* __restrict__ kb,
                          const float* __restrict__ vg,
                          const float* __restrict__ vb,
                          const float* __restrict__ lab,
                          float* __restrict__ gate,
                          float* __restrict__ Kb,
                          float* __restrict__ Vb,
                          int t) {
  const int n = blockIdx.x;
  const int tid = threadIdx.x; // == d
  __shared__ float xs[BB][FF];
  __shared__ float rsum[128], rsq[128];

  // stage x[:, t, n, :] into LDS (coalesced)
  for (int i = tid; i < BB * FF; i += 128) {
    const int b = i / FF, f = i % FF;
    xs[b][f] = x[(((size_t)b * TT + t) * NN + n) * FF + f];
  }
  __syncthreads();

  // input gate (one thread per batch)
  if (tid < BB) {
    float a = b_in[0];
#pragma unroll
    for (int f = 0; f < FF; ++f) a += xs[tid][f] * w_in[f];
    if (n < NN - 1) a += lab[tid * TT + t] * 0.1f;
    gate[tid * NN + n] = 1.f / (1.f + expf(-a));
  }

  // projections: thread d accumulates column d for all 8 batches
  float kacc[BB], vacc[BB];
#pragma unroll
  for (int b = 0; b < BB; ++b) { kacc[b] = 0.f; vacc[b] = 0.f; }
  const float* wkp = Wk + (size_t)n * FF * DD + tid;
  const float* wvp = Wv + (size_t)n * FF * DD + tid;
  for (int f = 0; f < FF; ++f) {
    const float wk = wkp[(size_t)f * DD];
    const float wv = wvp[(size_t)f * DD];
#pragma unroll
    for (int b = 0; b < BB; ++b) {
      kacc[b] += xs[b][f] * wk;
      vacc[b] += xs[b][f] * wv;
    }
  }

  const float bkd = bk[(size_t)n * DD + tid], bvd = bv[(size_t)n * DD + tid];
  const float kgd = kg[tid], kbd = kb[tid];
  const float vgd = vg[tid], vbd = vb[tid];
  const float inv_sqrt_d = 0.08838834764831845f; // 1/sqrt(128)

  for (int b = 0; b < BB; ++b) {
    // ---- K: LN over D then softplus * 1/sqrt(D)
    {
      const float val = kacc[b] + bkd;
      __syncthreads();
      rsum[tid] = val; rsq[tid] = val * val;
      __syncthreads();
      for (int s = 64; s > 0; s >>= 1) {
        if (tid < s) { rsum[tid] += rsum[tid + s]; rsq[tid] += rsq[tid + s]; }
        __syncthreads();
      }
      const float mean = rsum[0] * (1.f / DD);
      const float var  = rsq[0] * (1.f / DD) - mean * mean;
      const float yn = kgd * (val - mean) * rsqrtf(var + LN_EPS) + kbd;
      Kb[((size_t)b * NN + n) * DD + tid] = softplus_f(yn) * inv_sqrt_d;
    }
    // ---- V: LN over D then softplus
    {
      const float val = vacc[b] + bvd;
      __syncthreads();
      rsum[tid] = val; rsq[tid] = val * val;
      __syncthreads();
      for (int s = 64; s > 0; s >>= 1) {
        if (tid < s) { rsum[tid] += rsum[tid + s]; rsq[tid] += rsq[tid + s]; }
        __syncthreads();
      }
      const float mean = rsum[0] * (1.f / DD);
      const float var  = rsq[0] * (1.f / DD) - mean * mean;
      const float yn = vgd * (val - mean) * rsqrtf(var + LN_EPS) + vbd;
      Vb[((size_t)b * NN + n) * DD + tid] = softplus_f(yn);
    }
  }
}

// ---------------------------------------------------------------------------
// Stage 2 (per t): now_pre[b] = K[b] @ V[b]^T via V_WMMA_F32_16X16X4_F32.
// Block = 8 waves (256 thr).  Block covers 16 rows x 128 cols; each wave owns
// one 16x16 tile.  A tile staged in padded LDS; B fragments are per-lane b64
// loads of a consecutive d-pair from V (lane streams one V row -> cache hot).
//
// Fragment maps (ISA 7.12.2, wave32):
//   A 16x4 : lane (hi=lane>>4, lo=lane&15) holds row M=lo, K = d0+2*hi+{0,1}
//   B 4x16 : VGPR j holds K = d0+2*hi+j, col N = lo   (mirrors C half-split)
//   C 16x16: VGPR r holds row hi*8+r, col lo
// ---------------------------------------------------------------------------
__global__ void gemm_kernel(const float* __restrict__ Kb,
                            const float* __restrict__ Vb,
                            float* __restrict__ nowp) {
  __shared__ float At[16][132]; // padded to dodge bank conflicts on row reads
  const int b    = blockIdx.z;
  const int r0   = blockIdx.x * 16;
  const int mgrp = blockIdx.y * 128;
  const int tid  = threadIdx.x;
  const int wave = tid >> 5;
  const int lane = tid & 31;
  const int lo = lane & 15, hi = lane >> 4;
  const int m0 = mgrp + wave * 16;

  // cooperatively stage the 16x128 A tile (coalesced)
  const float* Kp = Kb + ((size_t)b * NN + r0) * DD;
  for (int i = tid; i < 16 * DD; i += 256) {
    At[i >> 7][i & 127] = Kp[(size_t)(i >> 7) * DD + (i & 127)];
  }
  __syncthreads();

  v8f acc = {};
  const float* Vp = Vb + ((size_t)b * NN + (m0 + lo)) * DD;
#pragma unroll
  for (int d0 = 0; d0 < DD; d0 += 4) {
    v2f a, bfr;
    a.x   = At[lo][d0 + 2 * hi];
    a.y   = At[lo][d0 + 2 * hi + 1];
    bfr.x = Vp[d0 + 2 * hi];
    bfr.y = Vp[d0 + 2 * hi + 1];
    acc = __builtin_amdgcn_wmma_f32_16x16x4_f32(
        /*neg_a=*/false, a, /*neg_b=*/false, bfr,
        /*c_mod=*/(short)0, acc, /*reuse_a=*/false, /*reuse_b=*/false);
  }

  // write C tile (rows hi*8+r, column lo within tile)
  float* outp = nowp + ((size_t)b * NN + r0 + hi * 8) * NN + (m0 + lo);
#pragma unroll
  for (int r = 0; r < 8; ++r) outp[(size_t)r * NN] = acc[r];
}

// ---------------------------------------------------------------------------
// Stage 3 (per t): per-row LN over 1024 cols, relu, gated cell blend.
// One block per (b,n) row, 256 threads, 4 elements per thread.
// ---------------------------------------------------------------------------
__global__ void ln_update_kernel(const float* __restrict__ nowp,
                                 const float* __restrict__ gate,
                                 const float* __restrict__ ng,
                                 const float* __restrict__ nb,
                                 float* __restrict__ cell) {
  const int row = blockIdx.x; // b*N + n
  const int tid = threadIdx.x;
  __shared__ float rsum[256], rsq[256];
  const float* rp = nowp + (size_t)row * NN;
  float y[4];
  float s = 0.f, s2 = 0.f;
#pragma unroll
  for (int j = 0; j < 4; ++j) {
    const float v = rp[tid + j * 256];
    y[j] = v; s += v; s2 += v * v;
  }
  rsum[tid] = s; rsq[tid] = s2;
  __syncthreads();
  for (int st = 128; st > 0; st >>= 1) {
    if (tid < st) { rsum[tid] += rsum[tid + st]; rsq[tid] += rsq[tid + st]; }
    __syncthreads();
  }
  const float mean = rsum[0] * (1.f / NN);
  const float var  = rsq[0] * (1.f / NN) - mean * mean;
  const float inv  = rsqrtf(var + LN_EPS);
  const float I    = gate[row];
  float* cp = cell + (size_t)row * NN;
#pragma unroll
  for (int j = 0; j < 4; ++j) {
    const int m = tid + j * 256;
    float yn = ng[m] * (y[j] - mean) * inv + nb[m];
    yn = fmaxf(yn, 0.f);
    cp[m] = (1.f - I) * cp[m] + I * yn;
  }
}

__global__ void zero_kernel(float* __restrict__ p, size_t n) {
  const size_t i = (size_t)blockIdx.x * blockDim.x + threadIdx.x;
  if (i < n) p[i] = 0.f;
}

// ---------------------------------------------------------------------------
extern "C" void kernel_launch(void* const* d_in, const int* in_sizes, int n_in,
                              void* d_out, int out_size, void* d_ws, size_t ws_size,
                              hipStream_t stream) {
  const float* x      = (const float*)d_in[0];
  const float* labels = (const float*)d_in[1];
  const float* Wk     = (const float*)d_in[2];
  const float* Wv     = (const float*)d_in[3];
  const float* bk     = (const float*)d_in[4];
  const float* bv     = (const float*)d_in[5];
  const float* w_in   = (const float*)d_in[6];
  const float* b_in   = (const float*)d_in[7];
  const float* kg     = (const float*)d_in[8];
  const float* kb     = (const float*)d_in[9];
  const float* vg     = (const float*)d_in[10];
  const float* vb     = (const float*)d_in[11];
  const float* ng     = (const float*)d_in[12];
  const float* nb     = (const float*)d_in[13];
  float* cell = (float*)d_out;

  // workspace layout (floats): lab | gate | K | V | now_pre  (~42 MB total)
  float* ws   = (float*)d_ws;
  float* lab  = ws;                                   // 128   (pad to 256)
  float* gate = ws + 256;                             // B*N = 8192
  float* Kbuf = ws + 256 + BB * NN;                   // B*N*D
  float* Vbuf = Kbuf + (size_t)BB * NN * DD;          // B*N*D
  float* nowp = Vbuf + (size_t)BB * NN * DD;          // B*N*N

  zero_kernel<<<(BB * NN * NN + 255) / 256, 256, 0, stream>>>(
      cell, (size_t)BB * NN * NN);
  lab_kernel<<<1, 128, 0, stream>>>(labels, lab);

  for (int t = 0; t < TT; ++t) {
    kv_kernel<<<NN, 128, 0, stream>>>(x, Wk, Wv, bk, bv, w_in, b_in,
                                      kg, kb, vg, vb, lab, gate, Kbuf, Vbuf, t);
    gemm_kernel<<<dim3(NN / 16, NN / 128, BB), 256, 0, stream>>>(Kbuf, Vbuf, nowp);
    ln_update_kernel<<<BB * NN, 256, 0, stream>>>(nowp, gate, ng, nb, cell);
  }
}